// TiramisuModelMult_4492535792502
// MI455X (gfx1250) — compile-verified
//
#include <hip/hip_runtime.h>
#include <math.h>

// ---------------------------------------------------------------------------
// Tiramisu tree-LSTM cost model, MI455X (gfx1250, wave32).
//  Pass 0: pad small weights (M->x16, K->x4, zero fill) so GEMMs are branch-free.
//  Kernel 1/1b: comp layer 1 (600x40960 matvec) -> bandwidth bound (98.3 MB),
//               split-K over 4800 blocks, deterministic tree reduction.
//  Kernel 2:    everything else in one persistent workgroup; all dense math
//               via V_WMMA_F32_16X16X4_F32 with MT-way multi-tile accumulation
//               (explicit B-operand reuse, MT independent wmma chains).
// ---------------------------------------------------------------------------

typedef __attribute__((ext_vector_type(2))) float v2f;
typedef __attribute__((ext_vector_type(8))) float v8f;

#define HD   180
#define HG   720          // 4*H
#define RLD  192          // padded ld for H-sized (180) GEMM outputs
#define RX   388          // padded rec-MLP input rows (386 -> 388)
#define BIGNUM 10000000000.0f

struct Args {
  // raw inputs
  const float *obs, *loops, *mask;
  const float *W1, *b1;
  const float *c_wih,*c_whh,*c_bih,*c_bhh;              // comps_lstm (720x180, no pad needed)
  const float *n_wih,*n_whh,*n_bih,*n_bhh;              // nodes_lstm
  const float *no_comps,*no_nodes;
  const int   *child,*comps,*has;
  float *out;
  // padded weights/biases (in workspace)
  const float *W2p,*b2p,*W3p,*b3p,*W4p,*b4p;
  const float *Wr1p,*br1p,*Wr2p,*br2p;
  const float *Wp1p,*bp1p,*Wp2p,*bp2p;
  const float *Wlp,*blp,*Wvp,*bvp;
  // activations (workspace)
  float *part1,*h1,*h2,*h3,*emb;
  float *X0,*X1,*Hs,*Cs,*Z,*compsh,*recout,*recx,*rtmp,*rt2,*nodesh,*f1,*f2,*lg,*lv;
};

__device__ __forceinline__ float sigf(float x) { return 1.0f / (1.0f + expf(-x)); }

// ---------------------------------------------------------------------------
// Branch-free fp32 WMMA GEMM on padded operands, MT M-tiles per wave-group:
//   C(Mp x 16*tN) = W(Mp x Kp, row-major, Mp%16==0, Kp%4==0)
//                   @ X(col-major, ld ldx even) [+C] [+bias(+bias2)] [relu]
// Each group: one vb load per k-step feeds MT wmmas (explicit B reuse, MT
// independent accumulate chains). Ragged M edge handled by clamping extra
// tiles onto the last valid tile (compute discarded at store, uniform branch).
// MT chosen per call so each wave runs its group loop at most once (no LICM
// hoisting/spill across tiles). Layouts per CDNA5 ISA 7.12.2:
//   A 16x4 : lanes 0-15 K={k0,k0+1}, lanes 16-31 K={k0+2,k0+3}, M = lane%16
//   B 4x16 : same K split, N = lane%16
//   C/D    : VGPR v -> rows m0+v (lanes 0-15) / m0+v+8 (lanes 16-31), N=lane%16
// ---------------------------------------------------------------------------
template <int MT, bool BIAS, bool BIAS2, bool RELU, bool ACCUM>
__device__ void gemm_wmma(const float* __restrict__ W, int Mp, int Kp,
                          const float* __restrict__ X, int ldx, int tN,
                          float* C, int ldc,
                          const float* __restrict__ bias0,
                          const float* __restrict__ bias1)
{
  const int lane   = threadIdx.x & 31;
  const int wave   = threadIdx.x >> 5;
  const int nwaves = blockDim.x >> 5;
  const int hi2    = (lane >= 16) ? 2 : 0;   // K sub-offset for hi half
  const int hi8    = (lane >= 16) ? 8 : 0;   // row sub-offset for C/D
  const int lm     = lane & 15;
  const int tM     = Mp >> 4;
  const int gM     = (tM + MT - 1) / MT;

  for (int g = wave; g < gM * tN; g += nwaves) {
    const int gm = g % gM;
    const int n0 = (g / gM) << 4;
    const float* xp = X + (size_t)(n0 + lm) * ldx + hi2;

    const float* wp[MT];
    float*       cp[MT];
    v8f          acc[MT];
    #pragma unroll
    for (int mt = 0; mt < MT; ++mt) {
      int ti = gm * MT + mt;
      if (ti > tM - 1) ti = tM - 1;          // clamp ragged edge
      const int m0 = ti << 4;
      wp[mt] = W + (size_t)(m0 + lm) * Kp + hi2;
      cp[mt] = C + (size_t)(n0 + lm) * ldc + m0 + hi8;
      v8f z = {0.f,0.f,0.f,0.f,0.f,0.f,0.f,0.f};
      if (ACCUM) {
        const float4 c0 = *(const float4*)(cp[mt]);
        const float4 c1 = *(const float4*)(cp[mt] + 4);
        z[0]=c0.x; z[1]=c0.y; z[2]=c0.z; z[3]=c0.w;
        z[4]=c1.x; z[5]=c1.y; z[6]=c1.z; z[7]=c1.w;
      }
      acc[mt] = z;
    }

    #pragma unroll 2
    for (int k0 = 0; k0 < Kp; k0 += 4) {
      const v2f vb = *(const v2f*)(xp + k0);
      #pragma unroll
      for (int mt = 0; mt < MT; ++mt) {
        const v2f va = *(const v2f*)(wp[mt] + k0);
        acc[mt] = __builtin_amdgcn_wmma_f32_16x16x4_f32(false, va, false, vb,
                                                        (short)0, acc[mt], false, false);
      }
    }

    #pragma unroll
    for (int mt = 0; mt < MT; ++mt) {
      if (gm * MT + mt < tM) {               // wave-uniform scalar branch
        const int m0 = ((gm * MT + mt) << 4) + hi8;
        v8f r = acc[mt];
        if (BIAS) {
          const float4 b0 = *(const float4*)(bias0 + m0);
          const float4 b1 = *(const float4*)(bias0 + m0 + 4);
          r[0]+=b0.x; r[1]+=b0.y; r[2]+=b0.z; r[3]+=b0.w;
          r[4]+=b1.x; r[5]+=b1.y; r[6]+=b1.z; r[7]+=b1.w;
        }
        if (BIAS2) {
          const float4 b0 = *(const float4*)(bias1 + m0);
          const float4 b1 = *(const float4*)(bias1 + m0 + 4);
          r[0]+=b0.x; r[1]+=b0.y; r[2]+=b0.z; r[3]+=b0.w;
          r[4]+=b1.x; r[5]+=b1.y; r[6]+=b1.z; r[7]+=b1.w;
        }
        if (RELU) {
          #pragma unroll
          for (int v = 0; v < 8; ++v) r[v] = fmaxf(r[v], 0.0f);
        }
        float4 s0, s1;
        s0.x=r[0]; s0.y=r[1]; s0.z=r[2]; s0.w=r[3];
        s1.x=r[4]; s1.y=r[5]; s1.z=r[6]; s1.w=r[7];
        *(float4*)(cp[mt])     = s0;
        *(float4*)(cp[mt] + 4) = s1;
      }
    }
  }
}

#define SYNC() do { __threadfence(); __syncthreads(); } while (0)

// One LSTM step for B batched columns (tN 16-col tiles). Xstep/Hs/Cs: HD x *,
// ld HD; Z: HG x *, ld HG.  z = Wih@x + Whh@h + bih + bhh; gates i,f,g,o.
// 720/16 = 45 M-tiles, MT=6 -> 8 groups per n-tile (<= 16 waves, 1 iter/wave).
__device__ void lstm_step(const float* wih, const float* whh,
                          const float* bih, const float* bhh,
                          const float* Xstep, int B,
                          float* Hs, float* Cs, float* Z)
{
  const int tN = (B + 15) >> 4;
  gemm_wmma<6,true,true,false,false>(wih, HG, HD, Xstep, HD, tN, Z, HG, bih, bhh); SYNC();
  gemm_wmma<6,false,false,false,true>(whh, HG, HD, Hs, HD, tN, Z, HG, nullptr, nullptr); SYNC();
  for (int idx = threadIdx.x; idx < HD * B; idx += blockDim.x) {
    const int n = idx / HD, j = idx % HD;
    const float* z = Z + n * HG;
    const float ig = sigf(z[j]);
    const float fg = sigf(z[HD + j]);
    const float gg = tanhf(z[2 * HD + j]);
    const float og = sigf(z[3 * HD + j]);
    const float c  = fg * Cs[n * HD + j] + ig * gg;
    Cs[n * HD + j] = c;
    Hs[n * HD + j] = og * tanhf(c);
  }
  SYNC();
}

// ---------------------------------------------------------------------------
// Pass 0: pad-copy a weight matrix (MxK -> MpxKp, zero fill) + its bias.
// ---------------------------------------------------------------------------
__global__ void k_pad(const float* __restrict__ W, const float* __restrict__ b,
                      int M, int K, int Mp, int Kp,
                      float* __restrict__ Wd, float* __restrict__ bd)
{
  const int total = Mp * Kp + Mp;
  for (int idx = blockIdx.x * blockDim.x + threadIdx.x; idx < total;
       idx += gridDim.x * blockDim.x) {
    if (idx < Mp * Kp) {
      const int m = idx / Kp, k = idx % Kp;
      Wd[idx] = (m < M && k < K) ? W[m * K + k] : 0.0f;
    } else {
      const int m = idx - Mp * Kp;
      bd[m] = (m < M) ? b[m] : 0.0f;
    }
  }
}

// ---------------------------------------------------------------------------
// Kernel 1: split-K partial dot products of comp layer 1 (bandwidth bound).
// grid = 600 rows * 8 chunks; each block: 256 threads * 5 float4 per thread.
// ---------------------------------------------------------------------------
__global__ __launch_bounds__(256) void k_dot_split(const float* __restrict__ W1,
                                                   const float* __restrict__ obs,
                                                   float* __restrict__ part)
{
  const int row = blockIdx.x >> 3;
  const int chunk = blockIdx.x & 7;
  const float4* w = (const float4*)(W1 + (size_t)row * 40960 + chunk * 5120);
  const float4* x = (const float4*)(obs + chunk * 5120);
  float s = 0.0f;
  #pragma unroll
  for (int it = 0; it < 5; ++it) {
    const int i = threadIdx.x + it * 256;
    const float4 a = w[i], b = x[i];
    s += a.x * b.x + a.y * b.y + a.z * b.z + a.w * b.w;
  }
  __shared__ float red[256];
  red[threadIdx.x] = s;
  __syncthreads();
  for (int off = 128; off > 0; off >>= 1) {
    if (threadIdx.x < off) red[threadIdx.x] += red[threadIdx.x + off];
    __syncthreads();
  }
  if (threadIdx.x == 0) part[blockIdx.x] = red[0];
}

__global__ void k_reduce1(const float* __restrict__ part,
                          const float* __restrict__ b1,
                          float* __restrict__ h1)
{
  const int r = blockIdx.x * blockDim.x + threadIdx.x;
  if (r < 600) {
    float s = b1[r];
    #pragma unroll
    for (int c = 0; c < 8; ++c) s += part[r * 8 + c];
    h1[r] = fmaxf(s, 0.0f);
  }
}

// ---------------------------------------------------------------------------
// Kernel 2: the whole latency chain in one workgroup (16 waves, wave32).
// ---------------------------------------------------------------------------
__global__ __launch_bounds__(512) void k_rest(Args a)
{
  const int tid = threadIdx.x, bs = blockDim.x;

  // comp MLP layers 2..4: 600->350->200->900 (padded 352/208/912)
  gemm_wmma<2,true,false,true,false>(a.W2p, 352, 600, a.h1, 600, 1, a.h2, 352, a.b2p, nullptr); SYNC();
  gemm_wmma<1,true,false,true,false>(a.W3p, 208, 352, a.h2, 352, 1, a.h3, 208, a.b3p, nullptr); SYNC();
  gemm_wmma<4,true,false,true,false>(a.W4p, 912, 200, a.h3, 208, 1, a.emb, 912, a.b4p, nullptr); SYNC();

  // comps LSTM for nodes 0..20 (batched, 2 steps); input indices come from the
  // last child's comps row (Python `ci` end-of-loop quirk).
  for (int idx = tid; idx < 21 * HD; idx += bs) {
    const int n = idx / HD, j = idx % HD;
    const int lc = a.child[n * 4 + 3];
    const int i0 = a.comps[lc * 5 + 0];
    const int i1 = a.comps[lc * 5 + 1];
    a.X0[n * HD + j] = a.emb[i0 * HD + j];
    a.X1[n * HD + j] = a.emb[i1 * HD + j];
    a.Hs[n * HD + j] = 0.0f;
    a.Cs[n * HD + j] = 0.0f;
  }
  SYNC();
  lstm_step(a.c_wih, a.c_whh, a.c_bih, a.c_bhh, a.X0, 21, a.Hs, a.Cs, a.Z);
  lstm_step(a.c_wih, a.c_whh, a.c_bih, a.c_bhh, a.X1, 21, a.Hs, a.Cs, a.Z);
  for (int idx = tid; idx < 21 * HD; idx += bs) {
    const int n = idx / HD, j = idx % HD;
    a.compsh[n * HD + j] = a.has[n] ? a.Hs[n * HD + j] : a.no_comps[j];
  }
  SYNC();

  // rec MLP for nodes 5..20 (batch 16): x = [no_nodes; comps_h; loop_feat]
  for (int idx = tid; idx < 16 * RX; idx += bs) {
    const int col = idx / RX, r = idx % RX, node = 5 + col;
    float v = 0.0f;
    if (r < HD)            v = a.no_nodes[r];
    else if (r < 2 * HD)   v = a.compsh[node * HD + (r - HD)];
    else if (r < 386)      v = a.loops[node * 26 + (r - 2 * HD)];
    a.recx[col * RX + r] = v;
  }
  SYNC();
  gemm_wmma<1,true,false,true,false>(a.Wr1p, 208, RX, a.recx, RX, 1, a.rtmp, 208, a.br1p, nullptr); SYNC();
  gemm_wmma<1,true,false,true,false>(a.Wr2p, RLD, 200, a.rtmp, 208, 1, a.rt2, RLD, a.br2p, nullptr); SYNC();
  for (int idx = tid; idx < 16 * HD; idx += bs) {
    const int col = idx / HD, j = idx % HD;
    a.recout[(5 + col) * RLD + j] = a.rt2[col * RLD + j];
  }
  SYNC();

  // nodes LSTM stage A: nodes 1..4 (batch 4), 4 steps over children's rec outputs
  for (int idx = tid; idx < 4 * HD; idx += bs) { a.Hs[idx] = 0.0f; a.Cs[idx] = 0.0f; }
  SYNC();
  for (int t = 0; t < 4; ++t) {
    for (int idx = tid; idx < 4 * HD; idx += bs) {
      const int i = idx / HD, j = idx % HD;
      const int c = a.child[(1 + i) * 4 + t];
      a.X0[i * HD + j] = a.recout[c * RLD + j];
    }
    SYNC();
    lstm_step(a.n_wih, a.n_whh, a.n_bih, a.n_bhh, a.X0, 4, a.Hs, a.Cs, a.Z);
  }
  for (int idx = tid; idx < 4 * HD; idx += bs)
    a.nodesh[HD + idx] = a.Hs[idx];              // nodesh cols 1..4
  SYNC();

  // rec MLP for nodes 1..4 (batch 4)
  for (int idx = tid; idx < 4 * RX; idx += bs) {
    const int col = idx / RX, r = idx % RX, node = 1 + col;
    float v = 0.0f;
    if (r < HD)            v = a.nodesh[node * HD + r];
    else if (r < 2 * HD)   v = a.compsh[node * HD + (r - HD)];
    else if (r < 386)      v = a.loops[node * 26 + (r - 2 * HD)];
    a.recx[col * RX + r] = v;
  }
  SYNC();
  gemm_wmma<1,true,false,true,false>(a.Wr1p, 208, RX, a.recx, RX, 1, a.rtmp, 208, a.br1p, nullptr); SYNC();
  gemm_wmma<1,true,false,true,false>(a.Wr2p, RLD, 200, a.rtmp, 208, 1, a.rt2, RLD, a.br2p, nullptr); SYNC();
  for (int idx = tid; idx < 4 * HD; idx += bs) {
    const int col = idx / HD, j = idx % HD;
    a.recout[(1 + col) * RLD + j] = a.rt2[col * RLD + j];
  }
  SYNC();

  // nodes LSTM stage B: node 0 (batch 1) over rec outputs of nodes 1..4
  for (int idx = tid; idx < HD; idx += bs) { a.Hs[idx] = 0.0f; a.Cs[idx] = 0.0f; }
  SYNC();
  for (int t = 0; t < 4; ++t) {
    for (int idx = tid; idx < HD; idx += bs) {
      const int c = a.child[t];
      a.X0[idx] = a.recout[c * RLD + idx];
    }
    SYNC();
    lstm_step(a.n_wih, a.n_whh, a.n_bih, a.n_bhh, a.X0, 1, a.Hs, a.Cs, a.Z);
  }
  for (int idx = tid; idx < HD; idx += bs) a.nodesh[idx] = a.Hs[idx];
  SYNC();

  // rec MLP for node 0 -> prog (recout col 0)
  for (int r = tid; r < RX; r += bs) {
    float v = 0.0f;
    if (r < HD)            v = a.nodesh[r];
    else if (r < 2 * HD)   v = a.compsh[r - HD];
    else if (r < 386)      v = a.loops[r - 2 * HD];
    a.recx[r] = v;
  }
  SYNC();
  gemm_wmma<1,true,false,true,false>(a.Wr1p, 208, RX, a.recx, RX, 1, a.rtmp, 208, a.br1p, nullptr); SYNC();
  gemm_wmma<1,true,false,true,false>(a.Wr2p, RLD, 200, a.rtmp, 208, 1, a.rt2, RLD, a.br2p, nullptr); SYNC();
  for (int idx = tid; idx < HD; idx += bs) a.recout[idx] = a.rt2[idx];
  SYNC();

  // pred MLP 180->200->180 (padded 208/192)
  gemm_wmma<1,true,false,true,false>(a.Wp1p, 208, HD, a.recout, RLD, 1, a.f1, 208, a.bp1p, nullptr); SYNC();
  gemm_wmma<1,true,false,true,false>(a.Wp2p, RLD, 200, a.f1, 208, 1, a.f2, RLD, a.bp2p, nullptr); SYNC();

  // logits (62 -> 64) and value (1 -> 16), no relu
  gemm_wmma<1,true,false,false,false>(a.Wlp, 64, HD, a.f2, RLD, 1, a.lg, 64, a.blp, nullptr); SYNC();
  gemm_wmma<1,true,false,false,false>(a.Wvp, 16, HD, a.f2, RLD, 1, a.lv, 16, a.bvp, nullptr); SYNC();

  for (int k = tid; k < 63; k += bs)
    a.out[k] = (k < 62) ? (a.lg[k] - BIGNUM * (1.0f - a.mask[k])) : a.lv[0];
}

// ---------------------------------------------------------------------------
// Host launcher. d_in order = setup_inputs() insertion order, params flattened
// in insertion order (comp W/b pairs, lstm dicts w_ih,w_hh,b_ih,b_hh, ...).
// Workspace use: ~3.2 MB (padded weights + activations).
// ---------------------------------------------------------------------------
extern "C" void kernel_launch(void* const* d_in, const int* in_sizes, int n_in,
                              void* d_out, int out_size, void* d_ws, size_t ws_size,
                              hipStream_t stream)
{
  (void)in_sizes; (void)out_size; (void)ws_size;
  if (n_in < 36) return;
  const float* W2  = (const float*)d_in[5];  const float* b2 = (const float*)d_in[6];
  const float* W3  = (const float*)d_in[7];  const float* b3 = (const float*)d_in[8];
  const float* W4  = (const float*)d_in[9];  const float* b4 = (const float*)d_in[10];
  const float* Wr1 = (const float*)d_in[21]; const float* br1 = (const float*)d_in[22];
  const float* Wr2 = (const float*)d_in[23]; const float* br2 = (const float*)d_in[24];
  const float* Wp1 = (const float*)d_in[25]; const float* bp1 = (const float*)d_in[26];
  const float* Wp2 = (const float*)d_in[27]; const float* bp2 = (const float*)d_in[28];
  const float* Wl  = (const float*)d_in[29]; const float* bl  = (const float*)d_in[30];
  const float* Wv  = (const float*)d_in[31]; const float* bv  = (const float*)d_in[32];

  Args a;
  a.obs   = (const float*)d_in[0];
  a.loops = (const float*)d_in[1];
  a.mask  = (const float*)d_in[2];
  a.W1 = (const float*)d_in[3];  a.b1 = (const float*)d_in[4];
  a.c_wih = (const float*)d_in[11]; a.c_whh = (const float*)d_in[12];
  a.c_bih = (const float*)d_in[13]; a.c_bhh = (const float*)d_in[14];
  a.n_wih = (const float*)d_in[15]; a.n_whh = (const float*)d_in[16];
  a.n_bih = (const float*)d_in[17]; a.n_bhh = (const float*)d_in[18];
  a.no_comps = (const float*)d_in[19];
  a.no_nodes = (const float*)d_in[20];
  a.child = (const int*)d_in[33];
  a.comps = (const int*)d_in[34];
  a.has   = (const int*)d_in[35];
  a.out   = (float*)d_out;

  // carve workspace (floats), 64-float aligned regions
  float* ws = (float*)d_ws;
  size_t o = 0;
  auto carve = [&](size_t n) { float* p = ws + o; o += (n + 63) & ~(size_t)63; return p; };
  // padded weights + biases
  float* W2p  = carve(352 * 600); float* b2p  = carve(352);
  float* W3p  = carve(208 * 352); float* b3p  = carve(208);
  float* W4p  = carve(912 * 200); float* b4p  = carve(912);
  float* Wr1p = carve(208 * RX);  float* br1p = carve(208);
  float* Wr2p = carve(RLD * 200); float* br2p = carve(RLD);
  float* Wp1p = carve(208 * HD);  float* bp1p = carve(208);
  float* Wp2p = carve(RLD * 200); float* bp2p = carve(RLD);
  float* Wlp  = carve(64 * HD);   float* blp  = carve(64);
  float* Wvp  = carve(16 * HD);   float* bvp  = carve(16);
  a.W2p=W2p; a.b2p=b2p; a.W3p=W3p; a.b3p=b3p; a.W4p=W4p; a.b4p=b4p;
  a.Wr1p=Wr1p; a.br1p=br1p; a.Wr2p=Wr2p; a.br2p=br2p;
  a.Wp1p=Wp1p; a.bp1p=bp1p; a.Wp2p=Wp2p; a.bp2p=bp2p;
  a.Wlp=Wlp; a.blp=blp; a.Wvp=Wvp; a.bvp=bvp;
  // activations; N=1 GEMM inputs sized for 16 garbage columns (column-isolated)
  a.part1  = carve(600 * 8);
  a.h1     = carve(600 * 16);
  a.h2     = carve(352 * 16);
  a.h3     = carve(208 * 16);
  a.emb    = carve(912 * 16);
  a.X0     = carve(HD * 32);
  a.X1     = carve(HD * 32);
  a.Hs     = carve(HD * 32);
  a.Cs     = carve(HD * 32);
  a.Z      = carve(HG * 32);
  a.compsh = carve(HD * 21);
  a.recout = carve(RLD * 21);
  a.recx   = carve(RX * 16);
  a.rtmp   = carve(208 * 16);
  a.rt2    = carve(RLD * 16);
  a.nodesh = carve(HD * 5);
  a.f1     = carve(208 * 16);
  a.f2     = carve(RLD * 16);
  a.lg     = carve(64 * 16);
  a.lv     = carve(16 * 16);

  // pass 0: pad small weights (total ~2.7 MB, trivial vs 98 MB layer 1)
  k_pad<<<128, 256, 0, stream>>>(W2,  b2,  350, 600, 352, 600, W2p,  b2p);
  k_pad<<<128, 256, 0, stream>>>(W3,  b3,  200, 350, 208, 352, W3p,  b3p);
  k_pad<<<128, 256, 0, stream>>>(W4,  b4,  900, 200, 912, 200, W4p,  b4p);
  k_pad<<<128, 256, 0, stream>>>(Wr1, br1, 200, 386, 208, RX,  Wr1p, br1p);
  k_pad<<<128, 256, 0, stream>>>(Wr2, br2, 180, 200, RLD, 200, Wr2p, br2p);
  k_pad<<<128, 256, 0, stream>>>(Wp1, bp1, 200, 180, 208, HD,  Wp1p, bp1p);
  k_pad<<<128, 256, 0, stream>>>(Wp2, bp2, 180, 200, RLD, 200, Wp2p, bp2p);
  k_pad<<<128, 256, 0, stream>>>(Wl,  bl,  62,  180, 64,  HD,  Wlp,  blp);
  k_pad<<<128, 256, 0, stream>>>(Wv,  bv,  1,   180, 16,  HD,  Wvp,  bvp);

  k_dot_split<<<4800, 256, 0, stream>>>(a.W1, a.obs, a.part1);
  k_reduce1<<<3, 256, 0, stream>>>(a.part1, a.b1, a.h1);
  k_rest<<<1, 512, 0, stream>>>(a);
}